// Decoder_60361470378447
// MI455X (gfx1250) — compile-verified
//
#include <hip/hip_runtime.h>
#include <math.h>

#define DD   2048   // hidden_d
#define VV   2048   // hidden_v
#define LSEQ 512
#define LP1  513    // L+1
#define PP   64
#define TS   65     // decode steps

typedef __attribute__((ext_vector_type(2))) float v2f;
typedef __attribute__((ext_vector_type(8))) float v8f;
typedef int v4i_vs __attribute__((vector_size(16)));   // matches builtin param type

#if defined(__AMDGCN__) && __has_builtin(__builtin_amdgcn_global_load_async_to_lds_b128)
#define HAVE_ASYNC_LDS 1
#else
#define HAVE_ASYNC_LDS 0
#endif

#if defined(__AMDGCN__) && __has_builtin(__builtin_amdgcn_s_wait_asynccnt)
#define WAIT_ASYNC(n) __builtin_amdgcn_s_wait_asynccnt(n)
#elif defined(__AMDGCN__)
#define WAIT_ASYNC(n) asm volatile("s_wait_asynccnt %0" ::"i"(n) : "memory")
#else
#define WAIT_ASYNC(n)
#endif

#if HAVE_ASYNC_LDS
// cache -> LDS, no VGPR round trip, tracked on ASYNCcnt
__device__ __forceinline__ void async_copy16(const float* g, float* l)
{
    __builtin_amdgcn_global_load_async_to_lds_b128(
        (__attribute__((address_space(1))) v4i_vs*)g,
        (__attribute__((address_space(3))) v4i_vs*)l,
        /*imm offset=*/0, /*cpol=*/0);
}
#endif

// ---------------------------------------------------------------------------
// enc_proj[m, n] = sum_k A[m, k] * W[n, k] + bias[n]
// A = encoder_states [513, 2048] row-major, W = We_w [2048, 2048] row-major.
// One wave per 32x32 output tile (2x2 register blocking, 4 v8f accumulators):
// each LDS operand quad {a0,a1,b0,b1} feeds 4 V_WMMA_F32_16X16X4_F32 ops.
// Staging uses GLOBAL_LOAD_ASYNC_TO_LDS_B128 with double-buffered LDS so the
// next K-chunk streams in while WMMAs consume the current one.
// ---------------------------------------------------------------------------
__global__ __launch_bounds__(32)
void encproj_wmma_kernel(const float* __restrict__ A,
                         const float* __restrict__ W,
                         const float* __restrict__ bias,
                         float* __restrict__ out)
{
    __shared__ float lA[2][32][68];   // padded stride to dodge bank conflicts
    __shared__ float lB[2][32][68];

    const int m0   = blockIdx.x * 32;          // 17 tiles (last partial)
    const int n0   = blockIdx.y * 32;          // 64 tiles
    const int lane = threadIdx.x;              // 0..31
    const int mi   = lane & 15;                // per-lane M (A) / N (B) index
    const int kb   = (lane >> 4) << 1;         // per-lane K sub-offset: 0 or 2

    // staging pattern: 2 consecutive rows per b128 issue -> 4 full cachelines
    const int srow = lane >> 4;                // 0 or 1
    const int scol = (lane & 15) * 4;          // float4 column

    v8f acc00 = {}, acc01 = {}, acc10 = {}, acc11 = {};

    // stage one 64-wide K chunk of A (clamped rows) and B into buffer `buf`
    auto stage = [&](int kc, int buf) {
#if HAVE_ASYNC_LDS
#pragma unroll
        for (int it = 0; it < 16; ++it) {
            int r  = 2 * it + srow;
            int gm = m0 + r; if (gm >= LP1) gm = LP1 - 1;
            async_copy16(A + (size_t)gm * DD + kc + scol, &lA[buf][r][scol]);
        }
#pragma unroll
        for (int it = 0; it < 16; ++it) {
            int r = 2 * it + srow;
            async_copy16(W + (size_t)(n0 + r) * DD + kc + scol, &lB[buf][r][scol]);
        }
#else
#pragma unroll
        for (int it = 0; it < 16; ++it) {
            int r  = 2 * it + srow;
            int gm = m0 + r; if (gm >= LP1) gm = LP1 - 1;
            *(float4*)(&lA[buf][r][scol]) =
                *(const float4*)(A + (size_t)gm * DD + kc + scol);
        }
#pragma unroll
        for (int it = 0; it < 16; ++it) {
            int r = 2 * it + srow;
            *(float4*)(&lB[buf][r][scol]) =
                *(const float4*)(W + (size_t)(n0 + r) * DD + kc + scol);
        }
#endif
    };

    stage(0, 0);                                // prologue: chunk 0 -> buf 0

    for (int kc = 0, buf = 0; kc < DD; kc += 64, buf ^= 1) {
        if (kc + 64 < DD) {
            stage(kc + 64, buf ^ 1);            // prefetch next chunk
            WAIT_ASYNC(32);                     // current chunk landed; prefetch may fly
        } else {
            WAIT_ASYNC(0);
        }
        __syncthreads();

#pragma unroll
        for (int k = 0; k < 64; k += 4) {
            v2f a0, a1, b0, b1;
            // A 16x4 layout: lanes 0-15 hold K={k,k+1}, lanes 16-31 K={k+2,k+3}
            a0.x = lA[buf][mi][k + kb];        a0.y = lA[buf][mi][k + kb + 1];
            a1.x = lA[buf][mi + 16][k + kb];   a1.y = lA[buf][mi + 16][k + kb + 1];
            // B 4x16 layout mirrors A's K striping; per-lane N = mi
            b0.x = lB[buf][mi][k + kb];        b0.y = lB[buf][mi][k + kb + 1];
            b1.x = lB[buf][mi + 16][k + kb];   b1.y = lB[buf][mi + 16][k + kb + 1];
            acc00 = __builtin_amdgcn_wmma_f32_16x16x4_f32(false, a0, false, b0,
                                                          (short)0, acc00, false, false);
            acc01 = __builtin_amdgcn_wmma_f32_16x16x4_f32(false, a0, false, b1,
                                                          (short)0, acc01, false, false);
            acc10 = __builtin_amdgcn_wmma_f32_16x16x4_f32(false, a1, false, b0,
                                                          (short)0, acc10, false, false);
            acc11 = __builtin_amdgcn_wmma_f32_16x16x4_f32(false, a1, false, b1,
                                                          (short)0, acc11, false, false);
        }
        __syncthreads();
    }

    // C/D layout: lane n = lane&15; VGPR g holds M = g + 8*(lane>>4)
    const int nA = n0 + mi;
    const int nB = nA + 16;
    const float bnA = bias[nA];
    const float bnB = bias[nB];
#pragma unroll
    for (int g = 0; g < 8; ++g) {
        int mTop = m0 + g + (lane >> 4) * 8;
        int mBot = mTop + 16;
        if (mTop < LP1) {
            out[(size_t)mTop * VV + nA] = acc00[g] + bnA;
            out[(size_t)mTop * VV + nB] = acc01[g] + bnB;
        }
        if (mBot < LP1) {
            out[(size_t)mBot * VV + nA] = acc10[g] + bnA;
            out[(size_t)mBot * VV + nB] = acc11[g] + bnB;
        }
    }
}

// ---------------------------------------------------------------------------
// h <- initial_state, c <- 0
// ---------------------------------------------------------------------------
__global__ __launch_bounds__(256)
void init_state_kernel(const float* __restrict__ initial_state,
                       float* __restrict__ h, float* __restrict__ c)
{
    int d = blockIdx.x * 256 + threadIdx.x;
    if (d < DD) { h[d] = initial_state[d]; c[d] = 0.0f; }
}

// ---------------------------------------------------------------------------
// preact[r] = W_hh[r,:] @ h + W_ih[r,:] @ x_t + b_ih[r] + b_hh[r]
// One wave per row; 32 lanes x 16 float4 each covers D=2048; b128 loads stream
// W_hh from L2 (entire 64 MiB weight stays L2-resident across steps).
// ---------------------------------------------------------------------------
__global__ __launch_bounds__(256)
void gates_kernel(const float* __restrict__ W_hh, const float* __restrict__ W_ih,
                  const float* __restrict__ b_ih, const float* __restrict__ b_hh,
                  const float* __restrict__ h,
                  const float* __restrict__ seq_points,
                  const int*   __restrict__ positions,
                  int step, float* __restrict__ preact)
{
    const int lane = threadIdx.x & 31;
    const int wave = threadIdx.x >> 5;
    const int row  = blockIdx.x * 8 + wave;          // 0..8191

    // teacher-forced input x_t: step 0 -> special (0,0,1), else stack_seq[pos]
    float x0 = 0.f, x1 = 0.f, x2 = 1.f;
    if (step > 0) {
        int p = positions[step - 1];
        if (p < LSEQ) { x0 = seq_points[p*3+0]; x1 = seq_points[p*3+1]; x2 = seq_points[p*3+2]; }
    }

    const float4* wr = (const float4*)(W_hh + (size_t)row * DD);
    const float4* hv = (const float4*)h;
    float sum = 0.f;
#pragma unroll
    for (int i = 0; i < 16; ++i) {
        float4 w  = wr[lane + i * 32];
        float4 hh = hv[lane + i * 32];
        sum = fmaf(w.x, hh.x, sum); sum = fmaf(w.y, hh.y, sum);
        sum = fmaf(w.z, hh.z, sum); sum = fmaf(w.w, hh.w, sum);
    }
#pragma unroll
    for (int off = 16; off > 0; off >>= 1) sum += __shfl_down(sum, off, 32);
    if (lane == 0) {
        preact[row] = sum + b_ih[row] + b_hh[row]
                    + W_ih[row*3+0]*x0 + W_ih[row*3+1]*x1 + W_ih[row*3+2]*x2;
    }
}

// ---------------------------------------------------------------------------
// LSTM pointwise: i,f,g,o -> c,h; also emit hidden_states[t]
// ---------------------------------------------------------------------------
__device__ __forceinline__ float sigmoidf_(float x) { return 1.0f / (1.0f + __expf(-x)); }

__global__ __launch_bounds__(256)
void lstm_update_kernel(const float* __restrict__ preact,
                        float* __restrict__ h, float* __restrict__ c,
                        float* __restrict__ hidden_out, int step)
{
    int d = blockIdx.x * 256 + threadIdx.x;
    if (d >= DD) return;
    float i = sigmoidf_(preact[d]);
    float f = sigmoidf_(preact[DD + d]);
    float g = tanhf(preact[2*DD + d]);
    float o = sigmoidf_(preact[3*DD + d]);
    float cn = f * c[d] + i * g;
    float hn = o * tanhf(cn);
    c[d] = cn; h[d] = hn;
    hidden_out[(size_t)step * DD + d] = hn;
}

// ---------------------------------------------------------------------------
// dproj[v] = Wd_w[v,:] @ h + Wd_b[v]   (wave per row, same scheme as gates)
// ---------------------------------------------------------------------------
__global__ __launch_bounds__(256)
void dproj_kernel(const float* __restrict__ Wd_w, const float* __restrict__ Wd_b,
                  const float* __restrict__ h, float* __restrict__ dproj)
{
    const int lane = threadIdx.x & 31;
    const int wave = threadIdx.x >> 5;
    const int row  = blockIdx.x * 8 + wave;          // 0..2047
    const float4* wr = (const float4*)(Wd_w + (size_t)row * DD);
    const float4* hv = (const float4*)h;
    float sum = 0.f;
#pragma unroll
    for (int i = 0; i < 16; ++i) {
        float4 w  = wr[lane + i * 32];
        float4 hh = hv[lane + i * 32];
        sum = fmaf(w.x, hh.x, sum); sum = fmaf(w.y, hh.y, sum);
        sum = fmaf(w.z, hh.z, sum); sum = fmaf(w.w, hh.w, sum);
    }
#pragma unroll
    for (int off = 16; off > 0; off >>= 1) sum += __shfl_down(sum, off, 32);
    if (lane == 0) dproj[row] = sum + Wd_b[row];
}

// ---------------------------------------------------------------------------
// logits[t, l] = sum_v v_w[v] * tanh(enc_proj[l, v] + dproj[v]) + v_b
// One block per l (513 blocks), 256 threads over V=2048.
// ---------------------------------------------------------------------------
__global__ __launch_bounds__(256)
void logits_kernel(const float* __restrict__ enc_proj, const float* __restrict__ dproj,
                   const float* __restrict__ v_w, const float* __restrict__ v_b,
                   float* __restrict__ logits_out, int step)
{
    __shared__ float red[8];
    const int l = blockIdx.x;
    const int tid = threadIdx.x;
    const float* ep = enc_proj + (size_t)l * VV;
    float s = 0.f;
#pragma unroll
    for (int i = 0; i < 8; ++i) {
        int v = tid + i * 256;
        s = fmaf(v_w[v], tanhf(ep[v] + dproj[v]), s);
    }
#pragma unroll
    for (int off = 16; off > 0; off >>= 1) s += __shfl_down(s, off, 32);
    if ((tid & 31) == 0) red[tid >> 5] = s;
    __syncthreads();
    if (tid == 0) {
        float t = 0.f;
#pragma unroll
        for (int i = 0; i < 8; ++i) t += red[i];
        logits_out[(size_t)step * LP1 + l] = t + v_b[0];
    }
}

// ---------------------------------------------------------------------------
// losses[t] = logsumexp(logits[t,:]) - logits[t, target]
// ---------------------------------------------------------------------------
__global__ __launch_bounds__(256)
void loss_kernel(const float* __restrict__ logits_row,
                 const int* __restrict__ positions,
                 float* __restrict__ loss_out, int step)
{
    __shared__ float redm[8];
    __shared__ float reds[8];
    const int tid = threadIdx.x;

    float m = -INFINITY;
    for (int i = tid; i < LP1; i += 256) m = fmaxf(m, logits_row[i]);
#pragma unroll
    for (int off = 16; off > 0; off >>= 1) m = fmaxf(m, __shfl_down(m, off, 32));
    if ((tid & 31) == 0) redm[tid >> 5] = m;
    __syncthreads();
    float mm = redm[0];
#pragma unroll
    for (int i = 1; i < 8; ++i) mm = fmaxf(mm, redm[i]);

    float s = 0.f;
    for (int i = tid; i < LP1; i += 256) s += __expf(logits_row[i] - mm);
#pragma unroll
    for (int off = 16; off > 0; off >>= 1) s += __shfl_down(s, off, 32);
    if ((tid & 31) == 0) reds[tid >> 5] = s;
    __syncthreads();
    if (tid == 0) {
        float ss = 0.f;
#pragma unroll
        for (int i = 0; i < 8; ++i) ss += reds[i];
        int tgt = (step < PP) ? positions[step] : LSEQ;
        loss_out[0] = logf(ss) + mm - logits_row[tgt];
    }
}

// ---------------------------------------------------------------------------
extern "C" void kernel_launch(void* const* d_in, const int* in_sizes, int n_in,
                              void* d_out, int out_size, void* d_ws, size_t ws_size,
                              hipStream_t stream)
{
    const float* initial_state  = (const float*)d_in[0];
    const float* encoder_states = (const float*)d_in[1];
    const float* seq_points     = (const float*)d_in[2];
    const int*   positions      = (const int*)d_in[3];
    const float* W_ih           = (const float*)d_in[4];
    const float* W_hh           = (const float*)d_in[5];
    const float* b_ih           = (const float*)d_in[6];
    const float* b_hh           = (const float*)d_in[7];
    const float* We_w           = (const float*)d_in[8];
    const float* We_b           = (const float*)d_in[9];
    const float* Wd_w           = (const float*)d_in[10];
    const float* Wd_b           = (const float*)d_in[11];
    const float* v_w            = (const float*)d_in[12];
    const float* v_b            = (const float*)d_in[13];

    // output tuple concat: logits [65,513], hidden [65,2048], losses [65]
    float* out_logits = (float*)d_out;
    float* out_hidden = out_logits + (size_t)TS * LP1;
    float* out_losses = out_hidden + (size_t)TS * DD;

    // workspace: enc_proj | preact | dproj | h | c  (~4.3 MB)
    float* ws       = (float*)d_ws;
    float* enc_proj = ws;                               // 513*2048
    float* preact   = enc_proj + (size_t)LP1 * VV;      // 8192
    float* dproj    = preact + 4 * DD;                  // 2048
    float* hbuf     = dproj + VV;                       // 2048
    float* cbuf     = hbuf + DD;                        // 2048

    // one-shot fp32 WMMA GEMM for the encoder projection (32x32 tile / wave)
    encproj_wmma_kernel<<<dim3(17, 64), 32, 0, stream>>>(encoder_states, We_w, We_b, enc_proj);
    init_state_kernel<<<8, 256, 0, stream>>>(initial_state, hbuf, cbuf);

    for (int t = 0; t < TS; ++t) {
        gates_kernel<<<1024, 256, 0, stream>>>(W_hh, W_ih, b_ih, b_hh, hbuf,
                                               seq_points, positions, t, preact);
        lstm_update_kernel<<<8, 256, 0, stream>>>(preact, hbuf, cbuf, out_hidden, t);
        dproj_kernel<<<256, 256, 0, stream>>>(Wd_w, Wd_b, hbuf, dproj);
        logits_kernel<<<LP1, 256, 0, stream>>>(enc_proj, dproj, v_w, v_b, out_logits, t);
        loss_kernel<<<1, 256, 0, stream>>>(out_logits + (size_t)t * LP1, positions,
                                           out_losses + t, t);
    }
}